// GraphSAGEModel_46686294507759
// MI455X (gfx1250) — compile-verified
//
#include <hip/hip_runtime.h>
#include <math.h>

#define N_NODES 100000
#define N_EDGES 1600000
#define KDIM 128
#define HID 128
#define ODIM 32
#define BN_EPS 1e-5f

typedef __attribute__((ext_vector_type(2))) float v2f;
typedef __attribute__((ext_vector_type(8))) float v8f;

// ---------------------------------------------------------------- utilities
__global__ void zero4_kernel(float4* __restrict__ p, int n4) {
    int i = blockIdx.x * blockDim.x + threadIdx.x;
    if (i < n4) p[i] = make_float4(0.f, 0.f, 0.f, 0.f);
}

__global__ void degree_kernel(const int* __restrict__ dst, float* __restrict__ deg) {
    int e = blockIdx.x * blockDim.x + threadIdx.x;
    if (e < N_EDGES) atomicAdd(&deg[dst[e]], 1.0f);
}

__global__ void invdeg_kernel(float* __restrict__ deg) {
    int i = blockIdx.x * blockDim.x + threadIdx.x;
    if (i < N_NODES) deg[i] = 1.0f / fmaxf(deg[i], 1.0f);
}

// one wave per edge; each lane moves 4 consecutive floats (coalesced 512B/edge)
__global__ void scatter_kernel(const float* __restrict__ h, const int* __restrict__ src,
                               const int* __restrict__ dst, float* __restrict__ agg) {
    int gid = blockIdx.x * blockDim.x + threadIdx.x;   // N_EDGES*32 threads
    int e   = gid >> 5;
    int f4  = (gid & 31) * 4;
    int s = src[e], d = dst[e];
    const float4 v = *(const float4*)(h + s * KDIM + f4);
    float* p = agg + d * KDIM + f4;
    atomicAdd(p + 0, v.x);
    atomicAdd(p + 1, v.y);
    atomicAdd(p + 2, v.z);
    atomicAdd(p + 3, v.w);
}

// ------------------------------------------------- fused SAGE GEMM (f32 WMMA)
// Y[m,n] = (relu?) ( X[m,:]@Ws + (agg[m,:]*inv[m])@Wn + b[n] )
// one 16x16 output tile per wave; K stepped by 4 via V_WMMA_F32_16X16X4_F32
template <int NOUT, bool RELU>
__global__ void sage_gemm_kernel(const float* __restrict__ X, const float* __restrict__ AGG,
                                 const float* __restrict__ inv, const float* __restrict__ Ws,
                                 const float* __restrict__ Wn, const float* __restrict__ bias,
                                 float* __restrict__ Y) {
    const int lane = threadIdx.x & 31;
    const int wave = threadIdx.x >> 5;
    const int tile = blockIdx.x * 4 + wave;          // grid sized exactly: no bounds check,
    const int TN   = NOUT / 16;                      // EXEC stays all-ones for WMMA
    const int tm   = tile / TN;
    const int tn   = tile % TN;
    const int mrow = tm * 16 + (lane & 15);          // A: lane&15 = M for both half-waves
    const int ncol = tn * 16 + (lane & 15);          // B/C/D: lane&15 = N
    const int khalf = (lane >> 4) * 2;               // lanes 16-31 hold K=2,3 of each step

    const float* xrow = X + mrow * KDIM;
    const float* arow = AGG + mrow * KDIM;
    const float  s    = inv[mrow];
    const float  bval = bias[ncol];

    v8f acc;
#pragma unroll
    for (int j = 0; j < 8; ++j) acc[j] = bval;       // bias is per-column -> replicate

#pragma unroll 8
    for (int k = 0; k < KDIM; k += 4) {
        v2f a_s = *reinterpret_cast<const v2f*>(xrow + k + khalf);   // (K, K+1) pair
        v2f a_n = *reinterpret_cast<const v2f*>(arow + k + khalf);
        a_n = a_n * s;                                               // fold inv_deg
        const float* wsp = Ws + (k + khalf) * NOUT + ncol;
        const float* wnp = Wn + (k + khalf) * NOUT + ncol;
        v2f b_s; b_s[0] = wsp[0]; b_s[1] = wsp[NOUT];
        v2f b_n; b_n[0] = wnp[0]; b_n[1] = wnp[NOUT];
        acc = __builtin_amdgcn_wmma_f32_16x16x4_f32(false, a_s, false, b_s,
                                                    (short)0, acc, false, false);
        acc = __builtin_amdgcn_wmma_f32_16x16x4_f32(false, a_n, false, b_n,
                                                    (short)0, acc, false, false);
    }

    // C/D layout: VGPR j -> M = j (lanes 0-15) or j+8 (lanes 16-31), N = lane&15
    const int mbase = tm * 16 + (lane >> 4) * 8;
#pragma unroll
    for (int j = 0; j < 8; ++j) {
        float v = acc[j];
        if (RELU) v = fmaxf(v, 0.0f);
        Y[(mbase + j) * NOUT + ncol] = v;
    }
}

// ------------------------------------------------------------- batch norm
__global__ void bn_stats_kernel(const float* __restrict__ h, float* __restrict__ stats) {
    const int f    = threadIdx.x & 127;
    const int half = threadIdx.x >> 7;                // 256 threads = 2 node-strides
    float sum = 0.f, sq = 0.f;
    for (int n = blockIdx.x * 2 + half; n < N_NODES; n += gridDim.x * 2) {
        float v = h[n * HID + f];
        sum += v; sq += v * v;
    }
    __shared__ float ls[256], lq[256];
    ls[threadIdx.x] = sum; lq[threadIdx.x] = sq;
    __syncthreads();
    if (threadIdx.x < 128) {
        atomicAdd(&stats[f],        ls[threadIdx.x] + ls[threadIdx.x + 128]);
        atomicAdd(&stats[128 + f],  lq[threadIdx.x] + lq[threadIdx.x + 128]);
    }
}

__global__ void bn_apply_kernel(float* __restrict__ h, const float* __restrict__ stats,
                                const float* __restrict__ gamma, const float* __restrict__ beta) {
    int gid = blockIdx.x * blockDim.x + threadIdx.x;  // N_NODES*HID threads, exact
    int f = gid & 127;
    float mu  = stats[f] * (1.0f / N_NODES);
    float var = stats[128 + f] * (1.0f / N_NODES) - mu * mu;
    float v = (h[gid] - mu) * rsqrtf(var + BN_EPS) * gamma[f] + beta[f];
    h[gid] = fmaxf(v, 0.0f);
}

// ----------------------------------------------------- readout: mean + softmax
__global__ void colsum_kernel(const float* __restrict__ h2, float* __restrict__ csum) {
    const int c = threadIdx.x & 31;
    const int g = threadIdx.x >> 5;                   // 8 row-groups per block
    float sum = 0.f;
    for (int n = blockIdx.x * 8 + g; n < N_NODES; n += gridDim.x * 8)
        sum += h2[n * ODIM + c];
    __shared__ float ls[256];
    ls[threadIdx.x] = sum;
    __syncthreads();
    if (threadIdx.x < 32) {
        float t = 0.f;
#pragma unroll
        for (int g2 = 0; g2 < 8; ++g2) t += ls[g2 * 32 + c];
        atomicAdd(&csum[c], t);
    }
}

__global__ void softmax_kernel(const float* __restrict__ csum, float* __restrict__ out) {
    int lane = threadIdx.x;                           // 32 threads, one wave
    float v = csum[lane] * (1.0f / N_NODES);
    float m = v;
    for (int off = 16; off > 0; off >>= 1) m = fmaxf(m, __shfl_xor(m, off, 32));
    float e = expf(v - m);
    float s = e;
    for (int off = 16; off > 0; off >>= 1) s += __shfl_xor(s, off, 32);
    out[lane] = e / s;
}

// ------------------------------------------------------------------ launcher
extern "C" void kernel_launch(void* const* d_in, const int* in_sizes, int n_in,
                              void* d_out, int out_size, void* d_ws, size_t ws_size,
                              hipStream_t stream) {
    (void)in_sizes; (void)n_in; (void)out_size; (void)ws_size;
    const float* feat = (const float*)d_in[0];
    const int*   src  = (const int*)d_in[1];
    const int*   dst  = (const int*)d_in[2];
    const float* Ws0 = (const float*)d_in[3],  *Wn0 = (const float*)d_in[4],  *b0 = (const float*)d_in[5];
    const float* Ws1 = (const float*)d_in[6],  *Wn1 = (const float*)d_in[7],  *b1 = (const float*)d_in[8];
    const float* Ws2 = (const float*)d_in[9],  *Wn2 = (const float*)d_in[10], *b2 = (const float*)d_in[11];
    const float* g0  = (const float*)d_in[12], *be0 = (const float*)d_in[13];
    const float* g1  = (const float*)d_in[14], *be1 = (const float*)d_in[15];
    float* out = (float*)d_out;

    char*  ws  = (char*)d_ws;
    size_t off = 0;
    auto alloc = [&](size_t bytes) -> void* {
        void* p = ws + off;
        off += (bytes + 255) & ~(size_t)255;
        return p;
    };
    float* inv   = (float*)alloc((size_t)N_NODES * 4);
    float* hA    = (float*)alloc((size_t)N_NODES * HID * 4);
    float* hB    = (float*)alloc((size_t)N_NODES * HID * 4);
    float* agg   = (float*)alloc((size_t)N_NODES * HID * 4);
    float* stats = (float*)alloc(256 * 4);
    float* csum  = (float*)alloc(32 * 4);

    const int B = 256;
    const int gemm_grid_h = (N_NODES / 16) * (HID / 16) / 4;   // 12500, exact
    const int gemm_grid_o = (N_NODES / 16) * (ODIM / 16) / 4;  // 3125, exact

    // degrees -> inv_deg (in place in `inv`)
    zero4_kernel<<<(N_NODES / 4 + B - 1) / B, B, 0, stream>>>((float4*)inv, N_NODES / 4);
    degree_kernel<<<(N_EDGES + B - 1) / B, B, 0, stream>>>(dst, inv);
    invdeg_kernel<<<(N_NODES + B - 1) / B, B, 0, stream>>>(inv);

    // ---- layer 0: feat -> hA (relu), then BN+relu in place
    zero4_kernel<<<N_NODES * HID / 4 / B, B, 0, stream>>>((float4*)agg, N_NODES * HID / 4);
    scatter_kernel<<<N_EDGES * 32 / B, B, 0, stream>>>(feat, src, dst, agg);
    sage_gemm_kernel<HID, true><<<gemm_grid_h, 128, 0, stream>>>(feat, agg, inv, Ws0, Wn0, b0, hA);
    zero4_kernel<<<1, 64, 0, stream>>>((float4*)stats, 64);
    bn_stats_kernel<<<256, 256, 0, stream>>>(hA, stats);
    bn_apply_kernel<<<N_NODES * HID / B, B, 0, stream>>>(hA, stats, g0, be0);

    // ---- layer 1: hA -> hB (relu), BN+relu in place
    zero4_kernel<<<N_NODES * HID / 4 / B, B, 0, stream>>>((float4*)agg, N_NODES * HID / 4);
    scatter_kernel<<<N_EDGES * 32 / B, B, 0, stream>>>(hA, src, dst, agg);
    sage_gemm_kernel<HID, true><<<gemm_grid_h, 128, 0, stream>>>(hA, agg, inv, Ws1, Wn1, b1, hB);
    zero4_kernel<<<1, 64, 0, stream>>>((float4*)stats, 64);
    bn_stats_kernel<<<256, 256, 0, stream>>>(hB, stats);
    bn_apply_kernel<<<N_NODES * HID / B, B, 0, stream>>>(hB, stats, g1, be1);

    // ---- layer 2: hB -> hA[:, :32] (no relu), then mean over nodes + softmax
    zero4_kernel<<<N_NODES * HID / 4 / B, B, 0, stream>>>((float4*)agg, N_NODES * HID / 4);
    scatter_kernel<<<N_EDGES * 32 / B, B, 0, stream>>>(hB, src, dst, agg);
    sage_gemm_kernel<ODIM, false><<<gemm_grid_o, 128, 0, stream>>>(hB, agg, inv, Ws2, Wn2, b2, hA);
    zero4_kernel<<<1, 8, 0, stream>>>((float4*)csum, 8);
    colsum_kernel<<<256, 256, 0, stream>>>(hA, csum);
    softmax_kernel<<<1, 32, 0, stream>>>(csum, out);
}